// _MultiCodebookQuantizer_76347338654343
// MI455X (gfx1250) — compile-verified
//
#include <hip/hip_runtime.h>
#include <hip/hip_bf16.h>

// ---------------- problem constants ----------------
#define N_ROWS   32768      // B*L = 64*512
#define D_FULL   512
#define NUM_CB   4
#define CB_SIZE  1024
#define SUB_D    128
#define K_STEPS  32         // SUB_D / 4 (K per wmma_f32_16x16x4_f32)
#define S_TILES  64         // CB_SIZE / 16
#define ROWS_PER_WAVE 16
#define WAVES_PER_BLK 8
#define ROWS_PER_BLK  128   // 8 waves * 16 rows
#define TILE_PITCH 132      // 128 + 4 dword pad -> conflict-free LDS banks
#define TILE_DW   (16*TILE_PITCH)

// output layout (flat concat of reference tuple, all as float)
#define OUT_IDX   0
#define OUT_ZQST  131072
#define OUT_ZQ    (131072 + 16777216)
#define OUT_SCAL  (131072 + 2*16777216)

typedef float v2f __attribute__((ext_vector_type(2)));
typedef float v8f __attribute__((ext_vector_type(8)));

// gfx1250 async memory->LDS copy (ASYNCcnt-tracked, no VGPR round-trip).
// Generic pointers to __shared__ carry the LDS byte offset in their low 32 bits
// (flat-aperture rule).
__device__ __forceinline__ void async_b128_to_lds(unsigned lds_addr, const float* gsrc)
{
    asm volatile("global_load_async_to_lds_b128 %0, %1, off"
                 :: "v"(lds_addr), "v"(gsrc)
                 : "memory");
}
__device__ __forceinline__ void wait_async0()
{
    asm volatile("s_wait_asynccnt 0" ::: "memory");
}

// ---------------- prep: code norms + zero accumulators ----------------
__global__ void vq_prep(const float* __restrict__ cb,
                        float* __restrict__ c2tab,
                        unsigned* __restrict__ counts,
                        double* __restrict__ lossp)
{
    const int tid = threadIdx.x;
    const int g = blockIdx.x * 256 + tid;
    if (g < NUM_CB * CB_SIZE) counts[g] = 0u;
    if (g == 0) *lossp = 0.0;

    const int code = blockIdx.x * WAVES_PER_BLK + (tid >> 5);   // flat (k,s)
    const int lane = tid & 31;
    const float4 v = *(const float4*)(cb + (size_t)code * SUB_D + lane * 4);
    float s = v.x*v.x + v.y*v.y + v.z*v.z + v.w*v.w;
#pragma unroll
    for (int off = 16; off > 0; off >>= 1) s += __shfl_xor(s, off, 32);
    if (lane == 0) c2tab[code] = s;
}

// ---------------- main: fused f32-WMMA GEMM + argmin + gather + loss ----------------
__global__ __launch_bounds__(256)
void vq_main(const float* __restrict__ z_e,
             const float* __restrict__ cb,
             const float* __restrict__ c2tab,
             unsigned* __restrict__ counts,
             double* __restrict__ lossp,
             float* __restrict__ out)
{
    __shared__ __align__(16) float sB[2 * TILE_DW];

    const int tid  = threadIdx.x;
    const int lane = tid & 31;
    const int wave = tid >> 5;
    const int col  = lane & 15;        // N / row-within-frag
    const int hi   = lane >> 4;        // half-wave
    const int k    = blockIdx.y;
    const int rowbase = blockIdx.x * ROWS_PER_BLK + wave * ROWS_PER_WAVE;
    const int myrow   = rowbase + col;

    // ---- A fragments: 16 rows x 128 f32, laid out for V_WMMA_F32_16X16X4_F32 ----
    // lane L (<16) row=L gets K = 4t+{0,1}; lane L+16 same row gets K = 4t+{2,3}
    const float* rp = z_e + (size_t)myrow * D_FULL + k * SUB_D + 2 * hi;
    v2f a[K_STEPS];
    float z2 = 0.f;
#pragma unroll
    for (int t = 0; t < K_STEPS; ++t) {
        a[t] = *(const v2f*)(rp + 4 * t);
        z2 = fmaf(a[t].x, a[t].x, fmaf(a[t].y, a[t].y, z2));
    }
    z2 += __shfl_xor(z2, 16, 32);      // full ||row||^2, row = col, on all lanes

    const float* cbk = cb    + (size_t)k * CB_SIZE * SUB_D;
    const float* c2k = c2tab + k * CB_SIZE;

    // staging: 512 float4 chunks per 16x128 tile, 2 per thread, async -> LDS
    const int ch0 = tid, ch1 = tid + 256;
    const int c0code = ch0 >> 5, c0d = (ch0 & 31) * 4;
    const int c1code = ch1 >> 5, c1d = (ch1 & 31) * 4;

    const unsigned lds0 = (unsigned)(uintptr_t)&sB[c0code * TILE_PITCH + c0d];
    const unsigned lds1 = (unsigned)(uintptr_t)&sB[c1code * TILE_PITCH + c1d];
    const unsigned bufBytes = TILE_DW * 4;

    // prologue: tile 0 -> buffer 0 (async)
    async_b128_to_lds(lds0, cbk + c0code * SUB_D + c0d);
    async_b128_to_lds(lds1, cbk + c1code * SUB_D + c1d);
    wait_async0();
    __syncthreads();

    float bestv[8];
    int   besti[8];
#pragma unroll
    for (int i = 0; i < 8; ++i) { bestv[i] = 3.4e38f; besti[i] = 0; }

    for (int st = 0; st < S_TILES; ++st) {
        // issue async prefetch of next B tile into the other buffer
        if (st + 1 < S_TILES) {
            const float*  src = cbk + (size_t)(st + 1) * 16 * SUB_D;
            const unsigned bo = ((st + 1) & 1) * bufBytes;
            async_b128_to_lds(lds0 + bo, src + c0code * SUB_D + c0d);
            async_b128_to_lds(lds1 + bo, src + c1code * SUB_D + c1d);
        }

        // compute 16x16 score tile from buffer st&1 (two interleaved acc chains)
        const float* bt = &sB[(st & 1) * TILE_DW + col * TILE_PITCH + 2 * hi];
        v8f acc0 = {0.f, 0.f, 0.f, 0.f, 0.f, 0.f, 0.f, 0.f};
        v8f acc1 = {0.f, 0.f, 0.f, 0.f, 0.f, 0.f, 0.f, 0.f};
#pragma unroll
        for (int t = 0; t < K_STEPS; t += 2) {
            v2f b0 = *(const v2f*)(bt + 4 * t);
            v2f b1 = *(const v2f*)(bt + 4 * t + 4);
            acc0 = __builtin_amdgcn_wmma_f32_16x16x4_f32(
                false, a[t],     false, b0, (short)0, acc0, false, false);
            acc1 = __builtin_amdgcn_wmma_f32_16x16x4_f32(
                false, a[t + 1], false, b1, (short)0, acc1, false, false);
        }

        // dist' = c2[s] - 2*dot  (z2 constant over s -> added later for loss only)
        const float c2s  = c2k[st * 16 + col];
        const int   cand = st * 16 + col;
#pragma unroll
        for (int vr = 0; vr < 8; ++vr) {
            float dot = acc0[vr] + acc1[vr];
            float v = fmaf(-2.f, dot, c2s);
            if (v < bestv[vr]) { bestv[vr] = v; besti[vr] = cand; }
        }

        wait_async0();          // this wave's async writes to LDS landed
        __syncthreads();        // all waves' writes visible before next compute
    }

    // cross-lane argmin within each 16-lane half (row = vr + 8*hi), first-index ties
#pragma unroll
    for (int vr = 0; vr < 8; ++vr) {
#pragma unroll
        for (int off = 8; off > 0; off >>= 1) {
            float ov = __shfl_xor(bestv[vr], off, 32);
            int   oi = __shfl_xor(besti[vr], off, 32);
            if (ov < bestv[vr] || (ov == bestv[vr] && oi < besti[vr])) {
                bestv[vr] = ov; besti[vr] = oi;
            }
        }
    }

    // loss: sum over 16 rows of  z2[row] + min(c2 - 2*dot)  ==  ||z - q||^2
    float lsum = 0.f;
#pragma unroll
    for (int r = 0; r < 16; ++r) {
        float bv = __shfl(bestv[r & 7], (r >> 3) << 4, 32);
        float zz = __shfl(z2, r, 32);
        lsum += zz + bv;
    }
    if (lane == 0) atomicAdd(lossp, (double)lsum);

    // outputs: indices, z_q_st, z_q_all, histogram
    float* outIdx = out + OUT_IDX;
    float* zqst   = out + OUT_ZQST;
    float* zqa    = out + OUT_ZQ;
#pragma unroll
    for (int r = 0; r < 16; ++r) {
        const int bi = __shfl(besti[r & 7], (r >> 3) << 4, 32);
        const int n  = rowbase + r;
        if (lane == 0) {
            outIdx[(size_t)n * NUM_CB + k] = (float)bi;
            atomicAdd(&counts[k * CB_SIZE + bi], 1u);
        }
        const float4 q = *(const float4*)(cbk + (size_t)bi * SUB_D + lane * 4);
        const size_t o = (size_t)n * D_FULL + k * SUB_D + lane * 4;
        *(float4*)(zqst + o) = q;
        *(float4*)(zqa  + o) = q;
    }
}

// ---------------- finalize: perplexity + scalar losses ----------------
__global__ void vq_final(const unsigned* __restrict__ counts,
                         const double* __restrict__ lossp,
                         float* __restrict__ out)
{
    __shared__ float perp[NUM_CB];
    const int tid = threadIdx.x, lane = tid & 31, k = tid >> 5;   // 4 waves
    float ent = 0.f;
#pragma unroll
    for (int j = 0; j < CB_SIZE / 32; ++j) {
        float p = (float)counts[k * CB_SIZE + lane + 32 * j] * (1.0f / (float)N_ROWS);
        ent += p * logf(fmaxf(p, 1e-8f));
    }
#pragma unroll
    for (int off = 16; off > 0; off >>= 1) ent += __shfl_xor(ent, off, 32);
    if (lane == 0) perp[k] = expf(-ent);
    __syncthreads();
    if (tid == 0) {
        float lm = (float)(*lossp * (1.0 / ((double)N_ROWS * (double)D_FULL)));
        out[OUT_SCAL + 0] = lm;   // codebook_loss
        out[OUT_SCAL + 1] = lm;   // commit_loss (identical forward value)
        out[OUT_SCAL + 2] = 0.25f * (perp[0] + perp[1] + perp[2] + perp[3]);
    }
}

// ---------------- launcher ----------------
extern "C" void kernel_launch(void* const* d_in, const int* in_sizes, int n_in,
                              void* d_out, int out_size, void* d_ws, size_t ws_size,
                              hipStream_t stream)
{
    const float* z_e = (const float*)d_in[0];   // (64,512,512) f32
    const float* cb  = (const float*)d_in[1];   // (4,1024,128) f32
    float* out = (float*)d_out;

    float*    c2tab  = (float*)d_ws;                       // 4096 f32
    unsigned* counts = (unsigned*)d_ws + 4096;             // 4096 u32
    double*   lossp  = (double*)((char*)d_ws + 32768);     // 1 f64

    vq_prep<<<512, 256, 0, stream>>>(cb, c2tab, counts, lossp);
    dim3 grid(N_ROWS / ROWS_PER_BLK, NUM_CB);
    vq_main<<<grid, 256, 0, stream>>>(z_e, cb, c2tab, counts, lossp, out);
    vq_final<<<1, 128, 0, stream>>>(counts, lossp, out);
}